// MambaG2G_30425548325255
// MI455X (gfx1250) — compile-verified
//
#include <hip/hip_runtime.h>
#include <math.h>

#define D_MODEL   1000
#define D_STATE   8
#define D_CONV    4
#define D_INNER   2000
#define DT_RANK   63
#define PROJ_W    (DT_RANK + 2*D_STATE)   /* 79 */
#define NPTS      1000
#define TSEQ      8
#define LIN_DIM   256
#define DIM_OUT   64

#define KC        64      /* K chunk staged in LDS */
#define KPAIRS    (KC/2)  /* K pairs per chunk */
#define BP        48      /* pair-row stride in v2f slots: 96 dwords = 32 banks -> conflict-free */

typedef __attribute__((ext_vector_type(2))) float v2f;
typedef __attribute__((ext_vector_type(8))) float v8f;

// ---------------------------------------------------------------------------
// Activation epilogue codes: 0=none 1=relu 2=elu(tanh(x)) 3=elu(x)+1+1e-14 4=softplus
__device__ __forceinline__ float act_apply(float v, int act) {
  switch (act) {
    case 1: return v > 0.0f ? v : 0.0f;
    case 2: { float t = tanhf(v); return t > 0.0f ? t : expm1f(t); }
    case 3: { float e = v > 0.0f ? v : expm1f(v); return e + 1.0f + 1e-14f; }
    case 4: return v > 20.0f ? v : log1pf(expf(v));
    default: return v;
  }
}

// ---------------------------------------------------------------------------
// FP32 WMMA GEMM with LDS-staged B (fragment-order float2 pairs) and 2x2
// register blocking.
//   C[M,N] = act(A op @ B + bias);  A: MxK (KxM if TRANSA), B: KxN row-major.
// Block = 256 threads = 8 waves -> 256(M) x 32(N) C tile.
// Wave  -> 32x32 sub-tile = 4 accumulators (independent WMMA chains).
// B pairs (B[2p][n], B[2p+1][n]) stored contiguously in LDS so each fragment
// read is one aligned ds_load_b64 straight into an even VGPR pair (no repack).
// K loop split into full chunks (guard-free; vectorized v2f A loads when legal)
// and one edge chunk (clamped A addresses + zero-filled B so products vanish).
// EXEC stays all-ones through every WMMA; no divergent exits before barriers.
template <bool TRANSA>
__global__ __launch_bounds__(256) void gemm_wmma_f32(
    const float* __restrict__ A, const float* __restrict__ B,
    const float* __restrict__ bias, float* __restrict__ C,
    int M, int N, int K, int lda, int ldb, int ldc, int act)
{
  __shared__ v2f Bs[KPAIRS * BP];

  const int Nblocks = (N + 31) >> 5;
  const int mb = blockIdx.x / Nblocks;
  const int nb = blockIdx.x - mb * Nblocks;
  const int blockM  = mb * 256;
  const int colBase = nb * 32;

  const int wv   = threadIdx.x >> 5;
  const int lane = threadIdx.x & 31;
  const int half = lane >> 4;       // 0: K pair (k,k+1); 1: (k+2,k+3)
  const int l16  = lane & 15;

  const int row0 = blockM + wv * 32 + l16;   // A fragment rows for the two M sub-tiles
  const int row1 = row0 + 16;
  const int r0c  = row0 < M ? row0 : M - 1;  // clamp for safe addressing
  const int r1c  = row1 < M ? row1 : M - 1;

  const int colL = colBase + l16;            // C columns (low / high 16)
  const int colH = colL + 16;

  // hoisted A row bases (row index for TRANSA, row pointer otherwise)
  const float* aRow0 = TRANSA ? (A + r0c) : (A + (long)r0c * lda);
  const float* aRow1 = TRANSA ? (A + r1c) : (A + (long)r1c * lda);
  const bool vecA = (!TRANSA) && ((lda & 1) == 0);   // aligned v2f loads legal

  v8f acc00 = {}, acc01 = {}, acc10 = {}, acc11 = {};

  const int kfull = (K / KC) * KC;
  int kc = 0;

  // ================= full chunks: no K guards anywhere =================
  for (; kc < kfull; kc += KC) {
    __syncthreads();
#pragma unroll
    for (int j = 0; j < (KPAIRS * 32) / 256; ++j) {
      int idx = threadIdx.x + j * 256;
      int pk  = idx >> 5;              // K-pair index within chunk
      int nl  = idx & 31;
      int gk  = kc + pk * 2;
      int gn  = colBase + nl;
      v2f v;
      v.x = (gn < N) ? B[(long)gk * ldb + gn]       : 0.0f;
      v.y = (gn < N) ? B[(long)(gk + 1) * ldb + gn] : 0.0f;
      Bs[pk * BP + nl] = v;
      // prefetch next K-chunk of B while this chunk computes
      if (gk + KC < K && gn < N)
        __builtin_prefetch(&B[(long)(gk + KC) * ldb + gn], 0, 1);
    }
    __syncthreads();
    if (vecA) {
      for (int kl = 0; kl < KC; kl += 4) {
        int g0 = kc + kl + half * 2;
        int p  = (kl >> 1) + half;                   // LDS pair-row
        v2f a0 = *(const v2f*)(aRow0 + g0);          // 8B-aligned: g0 even, lda even
        v2f a1 = *(const v2f*)(aRow1 + g0);
        v2f b0 = Bs[p * BP + l16];                   // single ds_load_b64 each
        v2f b1 = Bs[p * BP + 16 + l16];
        acc00 = __builtin_amdgcn_wmma_f32_16x16x4_f32(false, a0, false, b0, (short)0, acc00, false, false);
        acc01 = __builtin_amdgcn_wmma_f32_16x16x4_f32(false, a0, false, b1, (short)0, acc01, false, false);
        acc10 = __builtin_amdgcn_wmma_f32_16x16x4_f32(false, a1, false, b0, (short)0, acc10, false, false);
        acc11 = __builtin_amdgcn_wmma_f32_16x16x4_f32(false, a1, false, b1, (short)0, acc11, false, false);
      }
    } else {
      for (int kl = 0; kl < KC; kl += 4) {
        int g0 = kc + kl + half * 2;
        int p  = (kl >> 1) + half;
        v2f a0, a1;
        if (TRANSA) {
          a0.x = aRow0[(long)g0 * lda];  a0.y = aRow0[(long)(g0 + 1) * lda];
          a1.x = aRow1[(long)g0 * lda];  a1.y = aRow1[(long)(g0 + 1) * lda];
        } else {
          a0.x = aRow0[g0];  a0.y = aRow0[g0 + 1];
          a1.x = aRow1[g0];  a1.y = aRow1[g0 + 1];
        }
        v2f b0 = Bs[p * BP + l16];
        v2f b1 = Bs[p * BP + 16 + l16];
        acc00 = __builtin_amdgcn_wmma_f32_16x16x4_f32(false, a0, false, b0, (short)0, acc00, false, false);
        acc01 = __builtin_amdgcn_wmma_f32_16x16x4_f32(false, a0, false, b1, (short)0, acc01, false, false);
        acc10 = __builtin_amdgcn_wmma_f32_16x16x4_f32(false, a1, false, b0, (short)0, acc10, false, false);
        acc11 = __builtin_amdgcn_wmma_f32_16x16x4_f32(false, a1, false, b1, (short)0, acc11, false, false);
      }
    }
  }

  // ================= edge chunk: clamped A + zero-filled B =================
  if (kc < K) {
    __syncthreads();
#pragma unroll
    for (int j = 0; j < (KPAIRS * 32) / 256; ++j) {
      int idx = threadIdx.x + j * 256;
      int pk  = idx >> 5;
      int nl  = idx & 31;
      int gk  = kc + pk * 2;
      int gn  = colBase + nl;
      v2f v;
      v.x = (gk < K     && gn < N) ? B[(long)gk * ldb + gn]       : 0.0f;
      v.y = (gk + 1 < K && gn < N) ? B[(long)(gk + 1) * ldb + gn] : 0.0f;
      Bs[pk * BP + nl] = v;
    }
    __syncthreads();
    for (int kl = 0; kl < KC; kl += 4) {
      int g0  = kc + kl + half * 2;
      int p   = (kl >> 1) + half;
      int g0c = g0 < K ? g0 : K - 1;         // clamp: staged-zero B kills the product
      int g1c = g0 + 1 < K ? g0 + 1 : K - 1;
      v2f a0, a1;
      if (TRANSA) {
        a0.x = aRow0[(long)g0c * lda];  a0.y = aRow0[(long)g1c * lda];
        a1.x = aRow1[(long)g0c * lda];  a1.y = aRow1[(long)g1c * lda];
      } else {
        a0.x = aRow0[g0c];  a0.y = aRow0[g1c];
        a1.x = aRow1[g0c];  a1.y = aRow1[g1c];
      }
      v2f b0 = Bs[p * BP + l16];
      v2f b1 = Bs[p * BP + 16 + l16];
      acc00 = __builtin_amdgcn_wmma_f32_16x16x4_f32(false, a0, false, b0, (short)0, acc00, false, false);
      acc01 = __builtin_amdgcn_wmma_f32_16x16x4_f32(false, a0, false, b1, (short)0, acc01, false, false);
      acc10 = __builtin_amdgcn_wmma_f32_16x16x4_f32(false, a1, false, b0, (short)0, acc10, false, false);
      acc11 = __builtin_amdgcn_wmma_f32_16x16x4_f32(false, a1, false, b1, (short)0, acc11, false, false);
    }
  }

  // ---- epilogue: bias + activation, guarded stores ----
  float bvL = (bias && colL < N) ? bias[colL] : 0.0f;
  float bvH = (bias && colH < N) ? bias[colH] : 0.0f;
  const int mb0 = blockM + wv * 32 + half * 8;  // C layout: lanes0-15 M=i, lanes16-31 M=8+i
#pragma unroll
  for (int i = 0; i < 8; ++i) {
    int m0 = mb0 + i;
    int m1 = m0 + 16;
    if (m0 < M && colL < N) C[(long)m0 * ldc + colL] = act_apply(acc00[i] + bvL, act);
    if (m0 < M && colH < N) C[(long)m0 * ldc + colH] = act_apply(acc01[i] + bvH, act);
    if (m1 < M && colL < N) C[(long)m1 * ldc + colL] = act_apply(acc10[i] + bvL, act);
    if (m1 < M && colH < N) C[(long)m1 * ldc + colH] = act_apply(acc11[i] + bvH, act);
  }
}

// ---------------------------------------------------------------------------
// Elementwise / misc kernels
__global__ void k_transpose_in(const float* __restrict__ x_in, float* __restrict__ X, long total)
{
  long i = (long)blockIdx.x * blockDim.x + threadIdx.x;
  if (i >= total) return;
  int d = (int)(i % D_MODEL);
  long r = i / D_MODEL;
  int n = (int)(r % NPTS);
  int t = (int)(r / NPTS);
  X[i] = x_in[((long)n * TSEQ + t) * D_MODEL + d];
}

__global__ void k_relu_add_row(const float* __restrict__ X, const float* __restrict__ e1,
                               float* __restrict__ R, long total)
{
  long i = (long)blockIdx.x * blockDim.x + threadIdx.x;
  if (i >= total) return;
  float v = X[i] + e1[(int)(i % D_MODEL)];
  R[i] = v > 0.0f ? v : 0.0f;
}

__global__ void k_add_inplace(float* __restrict__ dst, const float* __restrict__ src, long total)
{
  long i = (long)blockIdx.x * blockDim.x + threadIdx.x;
  if (i >= total) return;
  dst[i] += src[i];
}

__global__ void k_bn_stats(const float* __restrict__ h, float* __restrict__ mu,
                           float* __restrict__ var, int total_td)
{
  int i = blockIdx.x * blockDim.x + threadIdx.x;
  if (i >= total_td) return;
  int d = i % D_MODEL;
  int t = i / D_MODEL;
  const float* p = h + (long)t * NPTS * D_MODEL + d;
  float s = 0.0f, s2 = 0.0f;
  for (int n = 0; n < NPTS; ++n) {
    float v = p[(long)n * D_MODEL];
    s += v; s2 += v * v;
  }
  float m = s * (1.0f / NPTS);
  mu[i]  = m;
  var[i] = s2 * (1.0f / NPTS) - m * m;
}

// z[n,t,d] = bn(h[t,n,d]) + out1[t,n,d]
__global__ void k_bn_apply_add_transpose(const float* __restrict__ h, const float* __restrict__ out1,
                                         const float* __restrict__ mu, const float* __restrict__ var,
                                         const float* __restrict__ g, const float* __restrict__ b,
                                         float* __restrict__ z, long total)
{
  long i = (long)blockIdx.x * blockDim.x + threadIdx.x;
  if (i >= total) return;
  int d = (int)(i % D_MODEL);
  long r = i / D_MODEL;
  int n = (int)(r % NPTS);
  int t = (int)(r / NPTS);
  int td = t * D_MODEL + d;
  float v = (h[i] - mu[td]) * rsqrtf(var[td] + 1e-5f) * g[d] + b[d] + out1[i];
  z[((long)n * TSEQ + t) * D_MODEL + d] = v;
}

// depthwise causal conv (D_CONV=4) over L + SiLU; x-part of xz (row stride 2*D_INNER)
__global__ void k_conv_silu(const float* __restrict__ xz, const float* __restrict__ convW,
                            const float* __restrict__ convB, float* __restrict__ xs,
                            int Bb, int L, long total)
{
  long i = (long)blockIdx.x * blockDim.x + threadIdx.x;
  if (i >= total) return;
  int e = (int)(i % D_INNER);
  long bl = i / D_INNER;
  int l = (int)(bl % L);
  int b = (int)(bl / L);
  float acc = convB[e];
#pragma unroll
  for (int k = 0; k < D_CONV; ++k) {
    int ls = l - (D_CONV - 1) + k;
    if (ls >= 0) acc += xz[((long)b * L + ls) * (2 * D_INNER) + e] * convW[e * D_CONV + k];
  }
  xs[i] = acc / (1.0f + expf(-acc));   // SiLU
}

// selective scan; y written in-place over xs; fused y = (acc + x*Dp) * silu(res)
__global__ void k_scan(const float* __restrict__ delta, const float* __restrict__ proj,
                       const float* __restrict__ xz, const float* __restrict__ A_log,
                       const float* __restrict__ Dp, float* __restrict__ xs_y,
                       int Bb, int L)
{
  int i = blockIdx.x * blockDim.x + threadIdx.x;
  if (i >= Bb * D_INNER) return;
  int e = i % D_INNER;
  int b = i / D_INNER;
  float Ae[D_STATE], h[D_STATE];
#pragma unroll
  for (int s = 0; s < D_STATE; ++s) {
    Ae[s] = -expf(A_log[e * D_STATE + s]);
    h[s] = 0.0f;
  }
  float dp = Dp[e];
  for (int l = 0; l < L; ++l) {
    long r = (long)b * L + l;
    float dt  = delta[r * D_INNER + e];
    float xt  = xs_y[r * D_INNER + e];
    float res = xz[r * (2 * D_INNER) + D_INNER + e];
    const float* p = proj + r * PROJ_W;
    float dx = dt * xt;
    float acc = 0.0f;
#pragma unroll
    for (int s = 0; s < D_STATE; ++s) {
      h[s] = expf(dt * Ae[s]) * h[s] + dx * p[DT_RANK + s];
      acc += h[s] * p[DT_RANK + D_STATE + s];
    }
    float y = acc + xt * dp;
    float sres = res / (1.0f + expf(-res));
    xs_y[r * D_INNER + e] = y * sres;
  }
}

__global__ void k_mean_L(const float* __restrict__ y, float* __restrict__ e, long total)
{
  long i = (long)blockIdx.x * blockDim.x + threadIdx.x;
  if (i >= total) return;
  int d = (int)(i % D_MODEL);
  int n = (int)(i / D_MODEL);
  float s = 0.0f;
#pragma unroll
  for (int l = 0; l < TSEQ; ++l) s += y[((long)n * TSEQ + l) * D_MODEL + d];
  e[i] = s * (1.0f / TSEQ);
}

// ---------------------------------------------------------------------------
// Host side
static inline int cdiv(long a, int b) { return (int)((a + b - 1) / b); }

static void launch_gemm(const float* A, const float* B, const float* bias, float* C,
                        int M, int N, int K, int lda, int ldb, int ldc, int act,
                        bool transA, hipStream_t s)
{
  int blocks = ((M + 255) / 256) * ((N + 31) / 32);
  if (transA)
    gemm_wmma_f32<true><<<blocks, 256, 0, s>>>(A, B, bias, C, M, N, K, lda, ldb, ldc, act);
  else
    gemm_wmma_f32<false><<<blocks, 256, 0, s>>>(A, B, bias, C, M, N, K, lda, ldb, ldc, act);
}

struct MambaP {
  const float *Win, *convW, *convB, *Wx, *Wdt, *bdt, *A_log, *Dp, *Wout;
};

static void run_mamba(const MambaP& p, const float* u, int Bb, int L,
                      float* xz, float* xs, float* proj, float* delta, float* out,
                      hipStream_t s)
{
  const int Mr = Bb * L;
  // u @ Win -> xz (Mr x 4000)
  launch_gemm(u, p.Win, nullptr, xz, Mr, 2 * D_INNER, D_MODEL,
              D_MODEL, 2 * D_INNER, 2 * D_INNER, 0, false, s);
  // depthwise conv + SiLU -> xs (Mr x 2000)
  {
    long total = (long)Mr * D_INNER;
    k_conv_silu<<<cdiv(total, 256), 256, 0, s>>>(xz, p.convW, p.convB, xs, Bb, L, total);
  }
  // xs @ Wx -> proj (Mr x 79)
  launch_gemm(xs, p.Wx, nullptr, proj, Mr, PROJ_W, D_INNER,
              D_INNER, PROJ_W, PROJ_W, 0, false, s);
  // softplus(proj[:, :63] @ Wdt + bdt) -> delta (Mr x 2000)
  launch_gemm(proj, p.Wdt, p.bdt, delta, Mr, D_INNER, DT_RANK,
              PROJ_W, D_INNER, D_INNER, 4, false, s);
  // scan (y in-place over xs)
  k_scan<<<cdiv((long)Bb * D_INNER, 256), 256, 0, s>>>(delta, proj, xz, p.A_log, p.Dp, xs, Bb, L);
  // y @ Wout -> out (Mr x 1000)
  launch_gemm(xs, p.Wout, nullptr, out, Mr, D_MODEL, D_INNER,
              D_INNER, D_MODEL, D_MODEL, 0, false, s);
}

extern "C" void kernel_launch(void* const* d_in, const int* in_sizes, int n_in,
                              void* d_out, int out_size, void* d_ws, size_t ws_size,
                              hipStream_t stream)
{
  // ---- resolve input index map (insertion order vs sorted pytree order) ----
  // logical order: 0 x_in, 1 emb, 2 gW1, 3 gb1, 4 gW2, 5 gb2, 6 bn_g, 7 bn_b,
  //   8..16 cm{Win,convW,convB,Wx,Wdt,bdt,A_log,Dp,Wout}, 17..25 mm{same},
  //   26 out_W, 27 out_b, 28 mu_W, 29 mu_b, 30 sg_W, 31 sg_b
  int I[32];
  const int sub[9] = {3, 8, 7, 5, 2, 6, 0, 1, 4};  // logical -> sorted-subdict offset
  if (n_in >= 32 && in_sizes[0] == NPTS * TSEQ * NPTS) {
    if (in_sizes[8] == D_MODEL * 2 * D_INNER) {
      for (int i = 0; i < 32; ++i) I[i] = i;                     // insertion / insertion
    } else {
      for (int i = 0; i < 8; ++i) I[i] = i;                      // insertion / sorted
      for (int j = 0; j < 9; ++j) { I[8 + j] = 8 + sub[j]; I[17 + j] = 17 + sub[j]; }
      for (int i = 26; i < 32; ++i) I[i] = i;
    }
  } else {                                                        // fully sorted pytree
    I[0] = 31; I[1] = 11; I[2] = 12; I[3] = 13; I[4] = 14; I[5] = 15; I[6] = 1; I[7] = 0;
    for (int j = 0; j < 9; ++j) { I[8 + j] = 2 + sub[j]; I[17 + j] = 16 + sub[j]; }
    I[26] = 27; I[27] = 28; I[28] = 25; I[29] = 26; I[30] = 29; I[31] = 30;
  }
  auto P = [&](int k) -> const float* { return (const float*)d_in[I[k]]; };

  const float* x_in = P(0);
  const float* emb  = P(1);
  const float* gW1  = P(2);  const float* gb1 = P(3);
  const float* gW2  = P(4);  const float* gb2 = P(5);
  const float* bn_g = P(6);  const float* bn_b = P(7);
  MambaP cm = {P(8),  P(9),  P(10), P(11), P(12), P(13), P(14), P(15), P(16)};
  MambaP mm = {P(17), P(18), P(19), P(20), P(21), P(22), P(23), P(24), P(25)};
  const float* out_W = P(26); const float* out_b = P(27);
  const float* mu_W  = P(28); const float* mu_b  = P(29);
  const float* sg_W  = P(30); const float* sg_b  = P(31);

  // ---- workspace layout (floats) ----
  float* ws = (float*)d_ws;
  const long SZ8  = 8L  * 1000 * 1000;  // 8e6
  float* o_X   = ws;                    // X (T,N,D)          8e6
  float* o_A   = ws + SZ8;              // R / t1 / out1,2    8e6
  float* o_B   = ws + 2 * SZ8;          // agg/S/g/h          8e6
  float* o_C   = ws + 3 * SZ8;          // xz                 32e6
  float* o_D   = ws + 3 * SZ8 + 32000000L;  // xs/y           16e6
  float* o_E   = o_D + 16000000L;       // proj               1e6
  float* o_F   = o_E + 1000000L;        // delta              16e6
  float* o_G   = o_F + 16000000L;       // z                  8e6
  float* o_mu  = o_G + SZ8;             // bn mean            8192
  float* o_var = o_mu + 8192;           // bn var             8192

  const long tot8 = SZ8;
  float* xo   = (float*)d_out;                 // (1000, 256)
  float* muO  = xo + NPTS * LIN_DIM;           // (1000, 64)
  float* sgO  = muO + NPTS * DIM_OUT;          // (1000, 64)

  // 1. X = transpose(x_in, (1,0,2))
  k_transpose_in<<<cdiv(tot8, 256), 256, 0, stream>>>(x_in, o_X, tot8);
  // 2. R = relu(X + emb[1])
  k_relu_add_row<<<cdiv(tot8, 256), 256, 0, stream>>>(o_X, emb + D_MODEL, o_A, tot8);
  // 3. agg[t] = X[t]^T @ R[t]
  for (int t = 0; t < TSEQ; ++t)
    launch_gemm(o_X + (long)t * 1000000, o_A + (long)t * 1000000, nullptr,
                o_B + (long)t * 1000000, D_MODEL, D_MODEL, NPTS,
                D_MODEL, D_MODEL, D_MODEL, 0, true, stream);
  // 4. S = X + agg (in place)
  k_add_inplace<<<cdiv(tot8, 256), 256, 0, stream>>>(o_B, o_X, tot8);
  // 5. t1 = relu(S @ gW1 + gb1)
  launch_gemm(o_B, gW1, gb1, o_A, TSEQ * NPTS, D_MODEL, D_MODEL,
              D_MODEL, D_MODEL, D_MODEL, 1, false, stream);
  // 6. g = t1 @ gW2 + gb2; h = g + X
  launch_gemm(o_A, gW2, gb2, o_B, TSEQ * NPTS, D_MODEL, D_MODEL,
              D_MODEL, D_MODEL, D_MODEL, 0, false, stream);
  k_add_inplace<<<cdiv(tot8, 256), 256, 0, stream>>>(o_B, o_X, tot8);
  // 7. batch-norm stats over axis=1
  k_bn_stats<<<cdiv(TSEQ * D_MODEL, 256), 256, 0, stream>>>(o_B, o_mu, o_var, TSEQ * D_MODEL);
  // 8. first mamba (cm) on X -> out1 in o_A
  run_mamba(cm, o_X, TSEQ, NPTS, o_C, o_D, o_E, o_F, o_A, stream);
  // 9. z = transpose(bn(h) + out1)
  k_bn_apply_add_transpose<<<cdiv(tot8, 256), 256, 0, stream>>>(
      o_B, o_A, o_mu, o_var, bn_g, bn_b, o_G, tot8);
  // 10. second mamba (mm) on z -> out2 in o_A
  run_mamba(mm, o_G, NPTS, TSEQ, o_C, o_D, o_E, o_F, o_A, stream);
  // 11. e = mean over L (into o_X, reused)
  k_mean_L<<<cdiv((long)NPTS * D_MODEL, 256), 256, 0, stream>>>(o_A, o_X, (long)NPTS * D_MODEL);
  // 12. xo = elu(tanh(e @ out_W + out_b))
  launch_gemm(o_X, out_W, out_b, xo, NPTS, LIN_DIM, D_MODEL,
              D_MODEL, LIN_DIM, LIN_DIM, 2, false, stream);
  // 13. mu = xo @ mu_W + mu_b
  launch_gemm(xo, mu_W, mu_b, muO, NPTS, DIM_OUT, LIN_DIM,
              LIN_DIM, DIM_OUT, DIM_OUT, 0, false, stream);
  // 14. sigma = elu(xo @ sg_W + sg_b) + 1 + 1e-14
  launch_gemm(xo, sg_W, sg_b, sgO, NPTS, DIM_OUT, LIN_DIM,
              LIN_DIM, DIM_OUT, DIM_OUT, 3, false, stream);
}